// MPCM_26886495273235
// MI455X (gfx1250) — compile-verified
//
#include <hip/hip_runtime.h>

#define HH   512
#define WW   512
#define NIMG 16
#define NPIX (HH * WW)
#define TILE 32
#define RMAX 13                   // max over k of (k + k/2) = 9 + 4
#define INW  (TILE + 2 * RMAX)    // 58
#define LSTR (INW + 1)            // 59 (odd -> LDS bank-conflict free)
#define TBX  (WW / TILE)          // 16
#define TBY  (HH / TILE)          // 16

// Probe-confirmed on this toolchain (round 1):
//   __builtin_amdgcn_global_load_async_to_lds_b32/b128   : YES
//   __builtin_amdgcn_global_store_async_from_lds_b32     : YES
//   __builtin_amdgcn_cluster_load_async_to_lds_b32       : YES
//   __builtin_amdgcn_s_wait_asynccnt / s_wait_tensorcnt  : YES
//   __builtin_amdgcn_tensor_load_to_lds                  : YES
// Builtin arg0 is a non-const global 'int*' (clang: "__device__ int *").
#if defined(__AMDGCN__) && __has_builtin(__builtin_amdgcn_global_load_async_to_lds_b32)
#  define USE_ASYNC 1
#else
#  define USE_ASYNC 0
#endif

#if USE_ASYNC
typedef __attribute__((address_space(1))) int GInt;   // global
typedef __attribute__((address_space(3))) int LInt;   // LDS

__device__ __forceinline__ void async_g2l_b32(const float* g, float* l) {
  // (global src, lds dst, imm offset, imm cpol)
  __builtin_amdgcn_global_load_async_to_lds_b32((GInt*)g, (LInt*)l, 0, 0);
}

__device__ __forceinline__ void wait_async_zero() {
#if __has_builtin(__builtin_amdgcn_s_wait_asynccnt)
  __builtin_amdgcn_s_wait_asynccnt(0);
#else
  asm volatile("s_wait_asynccnt 0" ::: "memory");
#endif
}
#endif

// =====================================================================
// Kernel 1: fused 4-scale box-mean + PCM + max, per 32x32 tile.
// Writes mpcm to `out` and per-block (sum, sumsq) partials for the
// per-image threshold. LDS: 3 * 58*59 fp32 tiles + 512 fp32 reduction.
// =====================================================================
__global__ __launch_bounds__(256)
void mpcm_tile_kernel(const float* __restrict__ x,
                      float* __restrict__ out,
                      float* __restrict__ partials)
{
  __shared__ float s_in[INW * LSTR];
  __shared__ float s_hs[INW * LSTR];
  __shared__ float s_bm[INW * LSTR];
  __shared__ float s_red[512];

  const int tid = threadIdx.x;
  const int bx = blockIdx.x, by = blockIdx.y, img = blockIdx.z;
  const int gx0 = bx * TILE - RMAX;
  const int gy0 = by * TILE - RMAX;
  const float* __restrict__ xi = x + (size_t)img * NPIX;

  // ---- stage 58x58 input tile in LDS (zeros outside the image) ----
#if USE_ASYNC
  {
    const bool border = (gx0 < 0) | (gy0 < 0) | (gx0 + INW > WW) | (gy0 + INW > HH);
    if (border) {
      for (int idx = tid; idx < INW * INW; idx += 256) {
        const int i = idx / INW, j = idx - i * INW;
        s_in[i * LSTR + j] = 0.0f;
      }
      __syncthreads();   // zero-fill visible before async writes overlay it
    }
    for (int idx = tid; idx < INW * INW; idx += 256) {
      const int i = idx / INW, j = idx - i * INW;
      const int gy = gy0 + i, gx = gx0 + j;
      if (((unsigned)gy < (unsigned)HH) & ((unsigned)gx < (unsigned)WW)) {
        async_g2l_b32(xi + (size_t)gy * WW + gx, &s_in[i * LSTR + j]);
      }
    }
    wait_async_zero();   // this wave's async LDS writes have landed
  }
#else
  for (int idx = tid; idx < INW * INW; idx += 256) {
    const int i = idx / INW, j = idx - i * INW;
    const int gy = gy0 + i, gx = gx0 + j;
    float v = 0.0f;
    if (((unsigned)gy < (unsigned)HH) & ((unsigned)gx < (unsigned)WW))
      v = xi[(size_t)gy * WW + gx];
    s_in[i * LSTR + j] = v;
  }
#endif
  __syncthreads();

  float acc[4];
  #pragma unroll
  for (int t = 0; t < 4; ++t) acc[t] = -3.402823466e38f;

  #pragma unroll
  for (int s = 0; s < 4; ++s) {
    const int k  = 3 + 2 * s;        // scale / dilation
    const int r  = k >> 1;           // box radius
    const int lo = RMAX - k;         // box-mean region origin (tile coords)
    const int sz = TILE + 2 * k;     // box-mean region size
    const float invk2 = 1.0f / (float)(k * k);
    const int nrows = sz + 2 * r;    // hsum rows needed
    const int nch   = (sz + 15) >> 4;

    if (s) __syncthreads();          // protect s_hs/s_bm reuse across scales

    // ---- horizontal sliding-window box sums ----
    for (int u = tid; u < nrows * nch; u += 256) {
      const int row = (lo - r) + (u % nrows);
      const int c0  = lo + (u / nrows) * 16;
      const int c1  = (c0 + 16 < lo + sz) ? (c0 + 16) : (lo + sz);
      const float* rin = &s_in[row * LSTR];
      float* rhs = &s_hs[row * LSTR];
      float ssum = 0.0f;
      #pragma unroll
      for (int t = -r; t <= r; ++t) ssum += rin[c0 + t];
      rhs[c0] = ssum;
      for (int c = c0 + 1; c < c1; ++c) {
        ssum += rin[c + r] - rin[c - r - 1];
        rhs[c] = ssum;
      }
    }
    __syncthreads();

    // ---- vertical sliding-window sums -> box mean (zero outside image) ----
    for (int u = tid; u < sz * nch; u += 256) {
      const int col = lo + (u % sz);
      const int rr0 = lo + (u / sz) * 16;
      const int rr1 = (rr0 + 16 < lo + sz) ? (rr0 + 16) : (lo + sz);
      const int gx = gx0 + col;
      const bool colin = ((unsigned)gx < (unsigned)WW);
      float ssum = 0.0f;
      #pragma unroll
      for (int t = -r; t <= r; ++t) ssum += s_hs[(rr0 + t) * LSTR + col];
      {
        const int gy = gy0 + rr0;
        s_bm[rr0 * LSTR + col] =
            (colin & ((unsigned)gy < (unsigned)HH)) ? ssum * invk2 : 0.0f;
      }
      for (int rr = rr0 + 1; rr < rr1; ++rr) {
        ssum += s_hs[(rr + r) * LSTR + col] - s_hs[(rr - r - 1) * LSTR + col];
        const int gy = gy0 + rr;
        s_bm[rr * LSTR + col] =
            (colin & ((unsigned)gy < (unsigned)HH)) ? ssum * invk2 : 0.0f;
      }
    }
    __syncthreads();

    // ---- PCM: 8 dilated diffs, pair products [0,4][1,5][2,6][3,7], min ----
    #pragma unroll
    for (int t = 0; t < 4; ++t) {
      const int p = tid + t * 256;
      const int i = RMAX + (p >> 5);
      const int j = RMAX + (p & 31);
      const float c  = s_bm[i * LSTR + j];
      const float d0 = c - s_bm[(i - k) * LSTR + (j - k)];
      const float d1 = c - s_bm[(i - k) * LSTR + j];
      const float d2 = c - s_bm[(i - k) * LSTR + (j + k)];
      const float d3 = c - s_bm[i * LSTR + (j - k)];
      const float d4 = c - s_bm[i * LSTR + (j + k)];
      const float d5 = c - s_bm[(i + k) * LSTR + (j - k)];
      const float d6 = c - s_bm[(i + k) * LSTR + j];
      const float d7 = c - s_bm[(i + k) * LSTR + (j + k)];
      const float m = fminf(fminf(d0 * d4, d1 * d5), fminf(d2 * d6, d3 * d7));
      acc[t] = fmaxf(acc[t], m);
    }
  }

  // ---- write mpcm tile + deterministic block reduction for mean/std ----
  float lsum = 0.0f, lss = 0.0f;
  float* __restrict__ oi = out + (size_t)img * NPIX;
  const int oy0 = by * TILE, ox0 = bx * TILE;
  #pragma unroll
  for (int t = 0; t < 4; ++t) {
    const int p = tid + t * 256;
    const int oy = oy0 + (p >> 5), ox = ox0 + (p & 31);
    const float v = acc[t];
    oi[(size_t)oy * WW + ox] = v;
    lsum += v;
    lss  += v * v;
  }
  s_red[tid]       = lsum;
  s_red[256 + tid] = lss;
  __syncthreads();
  for (int off = 128; off > 0; off >>= 1) {
    if (tid < off) {
      s_red[tid]       += s_red[tid + off];
      s_red[256 + tid] += s_red[256 + tid + off];
    }
    __syncthreads();
  }
  if (tid == 0) {
    const int bidx = (img * TBY + by) * TBX + bx;
    partials[2 * bidx]     = s_red[0];
    partials[2 * bidx + 1] = s_red[256];
  }
}

// =====================================================================
// Kernel 2: per-image reduce 256 partials -> threshold = mean + 3*std
// (unbiased, ddof=1). Deterministic fixed-order tree reduction.
// =====================================================================
__global__ __launch_bounds__(256)
void mpcm_thresh_kernel(const float* __restrict__ partials,
                        float* __restrict__ th)
{
  __shared__ float ssum[256];
  __shared__ float sss[256];
  const int img = blockIdx.x, tid = threadIdx.x;
  const int base = img * (TBX * TBY);
  ssum[tid] = partials[2 * (base + tid)];
  sss[tid]  = partials[2 * (base + tid) + 1];
  __syncthreads();
  for (int off = 128; off > 0; off >>= 1) {
    if (tid < off) { ssum[tid] += ssum[tid + off]; sss[tid] += sss[tid + off]; }
    __syncthreads();
  }
  if (tid == 0) {
    const float N = (float)NPIX;
    const float mean = ssum[0] / N;
    float var = (sss[0] - N * mean * mean) / (N - 1.0f);
    var = var > 0.0f ? var : 0.0f;
    th[img] = mean + 3.0f * sqrtf(var);
  }
}

// =====================================================================
// Kernel 3: in-place binarize d_out: out = (out > th[img]) ? 1 : 0
// =====================================================================
__global__ __launch_bounds__(256)
void mpcm_binarize_kernel(float4* __restrict__ out, const float* __restrict__ th)
{
  const int i = blockIdx.x * 256 + threadIdx.x;   // over NIMG*NPIX/4 elems
  const int img = i / (NPIX / 4);
  const float t = th[img];
  float4 v = out[i];
  v.x = (v.x > t) ? 1.0f : 0.0f;
  v.y = (v.y > t) ? 1.0f : 0.0f;
  v.z = (v.z > t) ? 1.0f : 0.0f;
  v.w = (v.w > t) ? 1.0f : 0.0f;
  out[i] = v;
}

extern "C" void kernel_launch(void* const* d_in, const int* in_sizes, int n_in,
                              void* d_out, int out_size, void* d_ws, size_t ws_size,
                              hipStream_t stream)
{
  (void)in_sizes; (void)n_in; (void)out_size; (void)ws_size;
  const float* x = (const float*)d_in[0];
  float* out = (float*)d_out;                  // mpcm intermediate, then binarized in place
  float* partials = (float*)d_ws;              // 2 * 4096 floats
  float* th = partials + 2 * NIMG * TBX * TBY; // 16 floats

  dim3 grid1(TBX, TBY, NIMG);
  mpcm_tile_kernel<<<grid1, dim3(256), 0, stream>>>(x, out, partials);
  mpcm_thresh_kernel<<<dim3(NIMG), dim3(256), 0, stream>>>(partials, th);
  mpcm_binarize_kernel<<<dim3((NIMG * NPIX / 4) / 256), dim3(256), 0, stream>>>(
      (float4*)out, th);
}